// EdgeNet_6622839570931
// MI455X (gfx1250) — compile-verified
//
#include <hip/hip_runtime.h>
#include <math.h>

#define N_NODES 8192
#define N_EDGES 16384
#define D_FEAT  64
#define NQ      8
#define PI_F    3.14159265358979323846f

typedef __attribute__((ext_vector_type(2))) float v2f;
typedef __attribute__((ext_vector_type(4))) float v4f;
typedef __attribute__((ext_vector_type(8))) float v8f;

// -------- workspace layout --------
// [0      , 64KB ) : src  int[16384]
// [64KB   , 128KB) : dst  int[16384]
// [128KB  , 640KB) : Z    float[16384*8]
// [640KB  , +8  )  : gmm  uint[2]  {min_bits, max_bits}

__global__ void init_ws(int* __restrict__ src, int* __restrict__ dst,
                        unsigned* __restrict__ gmm) {
    int i = blockIdx.x * blockDim.x + threadIdx.x;
    if (i < N_EDGES) { src[i] = 0; dst[i] = 0; }
    if (i == 0) { gmm[0] = 0x7F800000u; gmm[1] = 0u; }   // +inf, 0
}

// One streaming pass over the dense one-hot incidence matrices.
// src[e] = sum_n n * (Ro[n,e] > 0.5)  (exactly one nonzero per column).
// Each thread handles 4 consecutive edges (b128 load) over a 512-node chunk;
// partial sums combined with integer atomicAdd. Non-temporal loads: Ro/Ri are
// 512MB each and must not thrash the 192MB L2 (X and Z want to stay resident).
__global__ void extract_idx(const float* __restrict__ Ro,
                            const float* __restrict__ Ri,
                            int* __restrict__ src, int* __restrict__ dst) {
    const int g  = blockIdx.x * blockDim.x + threadIdx.x;   // float4 group
    const int n0 = blockIdx.y * 512;
    const v4f* M = (const v4f*)(blockIdx.z ? Ri : Ro);
    int* O = blockIdx.z ? dst : src;
    const int stride = N_EDGES / 4;
    int ax = 0, ay = 0, az = 0, aw = 0;
    #pragma unroll 4
    for (int n = n0; n < n0 + 512; ++n) {
        v4f v = __builtin_nontemporal_load(&M[(size_t)n * stride + g]);
        if (v.x > 0.5f) ax = n;
        if (v.y > 0.5f) ay = n;
        if (v.z > 0.5f) az = n;
        if (v.w > 0.5f) aw = n;
    }
    const int e = g * 4;
    if (ax) atomicAdd(&O[e + 0], ax);
    if (ay) atomicAdd(&O[e + 1], ay);
    if (az) atomicAdd(&O[e + 2], az);
    if (aw) atomicAdd(&O[e + 3], aw);
}

// Edge MLP: Z[e,q] = relu(concat(X[src[e]], X[dst[e]]) @ W_in + b_in)
// via V_WMMA_F32_16X16X4_F32. One wave per 16-edge tile, K=128 in 32 steps.
// Also produces global min/max of Z (for rescale01) via shuffle reduction +
// uint atomics (values are >= 0 post-relu, so float bits order as uints).
__global__ void edge_mlp_wmma(const float* __restrict__ X,
                              const float* __restrict__ W_in,
                              const float* __restrict__ b_in,
                              const int* __restrict__ src,
                              const int* __restrict__ dst,
                              float* __restrict__ Z,
                              unsigned* __restrict__ gmm) {
    const int lane = threadIdx.x & 31;
    const int wid  = threadIdx.x >> 5;
    const int tile = blockIdx.x * 8 + wid;
    const int e0   = tile * 16;
    const int row  = lane & 15;      // A: M row / B,C: N column
    const int half = lane >> 4;      // selects K sub-pair and C row half

    const int sidx = src[e0 + row];
    const int didx = dst[e0 + row];
    const float* __restrict__ rowS = X + (size_t)sidx * D_FEAT;
    const float* __restrict__ rowD = X + (size_t)didx * D_FEAT;

    v8f c = {};
    #pragma unroll
    for (int s = 0; s < 32; ++s) {
        const int kk = 4 * s + 2 * half;            // this lane's K pair
        const float* ap = (kk < D_FEAT) ? (rowS + kk) : (rowD + (kk - D_FEAT));
        v2f a; a.x = ap[0]; a.y = ap[1];
        v2f b;
        if (row < NQ) { b.x = W_in[kk * NQ + row]; b.y = W_in[(kk + 1) * NQ + row]; }
        else          { b.x = 0.0f;                b.y = 0.0f; }
        c = __builtin_amdgcn_wmma_f32_16x16x4_f32(
                /*neg_a=*/false, a, /*neg_b=*/false, b,
                /*c_mod=*/(short)0, c, /*reuse_a=*/false, /*reuse_b=*/false);
    }

    float mn = __int_as_float(0x7F800000);  // +inf
    float mx = 0.0f;                        // relu output >= 0
    if (row < NQ) {
        const float bias = b_in[row];
        #pragma unroll
        for (int j = 0; j < 8; ++j) {
            const int m = half * 8 + j;             // C row -> edge within tile
            float v = c[j] + bias;
            v = v > 0.0f ? v : 0.0f;
            Z[(size_t)(e0 + m) * NQ + row] = v;
            mn = v < mn ? v : mn;
            mx = v > mx ? v : mx;
        }
    }
    #pragma unroll
    for (int off = 16; off >= 1; off >>= 1) {
        float omn = __shfl_xor(mn, off, 32);
        float omx = __shfl_xor(mx, off, 32);
        mn = omn < mn ? omn : mn;
        mx = omx > mx ? omx : mx;
    }
    if (lane == 0) {
        atomicMin(&gmm[0], __float_as_uint(mn));
        atomicMax(&gmm[1], __float_as_uint(mx));
    }
}

// out[e] = sigmoid( sum_q cos((Z[e,q]-lo)/(hi-lo)*pi + theta_q[q]) * W_out[q] + b_out )
__global__ void finalize_k(const float* __restrict__ Z,
                           const float* __restrict__ theta,
                           const float* __restrict__ W_out,
                           const float* __restrict__ b_out,
                           const unsigned* __restrict__ gmm,
                           float* __restrict__ out) {
    const int e = blockIdx.x * blockDim.x + threadIdx.x;
    if (e >= N_EDGES) return;
    const float lo = __uint_as_float(gmm[0]);
    const float hi = __uint_as_float(gmm[1]);
    const float sc = PI_F / (hi - lo);
    float acc = b_out[0];
    #pragma unroll
    for (int q = 0; q < NQ; ++q) {
        const float h = (Z[(size_t)e * NQ + q] - lo) * sc;
        acc += cosf(h + theta[q] + theta[q + NQ]) * W_out[q];
    }
    out[e] = 1.0f / (1.0f + expf(-acc));
}

extern "C" void kernel_launch(void* const* d_in, const int* in_sizes, int n_in,
                              void* d_out, int out_size, void* d_ws, size_t ws_size,
                              hipStream_t stream) {
    const float* X     = (const float*)d_in[0];
    const float* Ri    = (const float*)d_in[1];
    const float* Ro    = (const float*)d_in[2];
    const float* W_in  = (const float*)d_in[3];
    const float* b_in  = (const float*)d_in[4];
    const float* theta = (const float*)d_in[5];
    const float* W_out = (const float*)d_in[6];
    const float* b_out = (const float*)d_in[7];
    float* out = (float*)d_out;

    char* ws = (char*)d_ws;
    int*      src = (int*)ws;
    int*      dst = (int*)(ws + 64 * 1024);
    float*    Z   = (float*)(ws + 128 * 1024);
    unsigned* gmm = (unsigned*)(ws + 640 * 1024);

    init_ws<<<64, 256, 0, stream>>>(src, dst, gmm);                 // 16384 threads
    dim3 gExt(16, 16, 2);                                           // edges x nchunk x {Ro,Ri}
    extract_idx<<<gExt, 256, 0, stream>>>(Ro, Ri, src, dst);        // the 1GB streaming pass
    edge_mlp_wmma<<<128, 256, 0, stream>>>(X, W_in, b_in, src, dst, Z, gmm);
    finalize_k<<<64, 256, 0, stream>>>(Z, theta, W_out, b_out, gmm, out);
}